// GIN__4698694222355
// MI455X (gfx1250) — compile-verified
//
#include <hip/hip_runtime.h>
#include <math.h>

#define NNODES 100000

typedef __attribute__((ext_vector_type(2))) float v2f;
typedef __attribute__((ext_vector_type(8))) float v8f;

// ---------------------------------------------------------------------------
// Edge scatter-add: out[dst[e]*F + f] += feat[src[e]*F + f], F = 1<<SHIFT.
// One thread per (edge, feature); consecutive lanes = consecutive features
// so gather reads and atomic adds are coalesced. Feature buffers (25-51 MB)
// live in the 192MB L2, so the fp32 atomics resolve in L2.
// ---------------------------------------------------------------------------
template <int SHIFT>
__global__ void scatter_add_kernel(const float* __restrict__ feat,
                                   float* __restrict__ out,
                                   const int* __restrict__ src,
                                   const int* __restrict__ dst,
                                   long long nthreads) {
  long long tid = (long long)blockIdx.x * blockDim.x + threadIdx.x;
  if (tid >= nthreads) return;
  int f = (int)(tid & ((1 << SHIFT) - 1));
  long long e = tid >> SHIFT;
  long long s = src[e];
  long long d = dst[e];
  atomicAdd(&out[(d << SHIFT) + f], feat[(s << SHIFT) + f]);
}

// ---------------------------------------------------------------------------
// fp32 WMMA GEMM:  C[M x N] = act(A[M x K] @ W[K x N] + bias)
// One wave32 per 16x16 output tile, V_WMMA_F32_16X16X4_F32 over K in steps
// of 4, fully unrolled (K, N compile-time). Fragment layouts (ISA 7.12.2):
//   A 16x4 : lanes 0-15 rows M, v0/v1 = K0/K1; lanes 16-31 = K2/K3
//   B 4x16 : lanes 0-15 cols N, v0/v1 = K0/K1; lanes 16-31 = K2/K3
//   C 16x16: VGPR r -> row r (lanes 0-15) / row r+8 (lanes 16-31), lane = col
// For N not a multiple of 16, OOB columns are handled by CLAMPING the B-load
// address (B col n only feeds output col n, owned by the same lane, which is
// simply not stored) — so the unrolled loop is completely branch-free.
// ---------------------------------------------------------------------------
template <int K, int N, int RELU>
__global__ void __launch_bounds__(32)
gemm_wmma_f32(const float* __restrict__ A, const float* __restrict__ W,
              const float* __restrict__ bias, float* __restrict__ C) {
  const int lane = threadIdx.x & 31;
  const int m    = lane & 15;     // row within A-frag / col within B,C frag
  const int hi   = lane >> 4;     // K sub-pair selector
  const int col0 = blockIdx.x << 4;
  const int row0 = blockIdx.y << 4;
  const int n    = col0 + m;
  const bool nin = ((N & 15) == 0) || (n < N);
  const int  nld = nin ? n : (N - 1);  // clamped load column

  const float* arow = A + (size_t)(row0 + m) * K + (hi << 1);
  const float* wcol = W + (size_t)(hi << 1) * N + nld;

  v8f acc = {};
#pragma unroll
  for (int k0 = 0; k0 < K; k0 += 4) {
    v2f a = *(const v2f*)(arow + k0);             // 8B-aligned: K even
    v2f b;
    b.x = wcol[(size_t)k0 * N];
    b.y = wcol[(size_t)(k0 + 1) * N];
    // (neg_a, A, neg_b, B, c_mod, C, reuse_a, reuse_b)
    acc = __builtin_amdgcn_wmma_f32_16x16x4_f32(false, a, false, b,
                                                (short)0, acc, false, false);
  }

  if (!nin) return;  // divergence only after all WMMAs have issued
  const float bv = bias[n];
#pragma unroll
  for (int r = 0; r < 8; ++r) {
    float v = acc[r] + bv;
    if (RELU) v = fmaxf(v, 0.0f);
    C[(size_t)(row0 + r + (hi << 3)) * N + n] = v;
  }
}

// ---------------------------------------------------------------------------
// Row-wise log_softmax over 40 logits. One wave32 per row:
// lane holds logit[lane], lanes 0..7 additionally logit[lane+32].
// ---------------------------------------------------------------------------
__global__ void log_softmax40_kernel(const float* __restrict__ in,
                                     float* __restrict__ out, int M) {
  int gtid = blockIdx.x * blockDim.x + threadIdx.x;
  int row  = gtid >> 5;
  int lane = gtid & 31;
  if (row >= M) return;
  const float* r = in + (size_t)row * 40;
  float a = r[lane];
  float b = (lane < 8) ? r[lane + 32] : -INFINITY;

  float mx = fmaxf(a, b);
  for (int off = 16; off > 0; off >>= 1)
    mx = fmaxf(mx, __shfl_xor(mx, off, 32));

  float s = expf(a - mx) + ((lane < 8) ? expf(b - mx) : 0.0f);
  for (int off = 16; off > 0; off >>= 1)
    s += __shfl_xor(s, off, 32);

  float lse = mx + logf(s);
  out[(size_t)row * 40 + lane] = a - lse;
  if (lane < 8) out[(size_t)row * 40 + lane + 32] = b - lse;
}

// ---------------------------------------------------------------------------
extern "C" void kernel_launch(void* const* d_in, const int* in_sizes, int n_in,
                              void* d_out, int out_size, void* d_ws, size_t ws_size,
                              hipStream_t stream) {
  const float* x    = (const float*)d_in[0];
  const int*   eidx = (const int*)d_in[1];
  const float* W1a  = (const float*)d_in[2];
  const float* b1a  = (const float*)d_in[3];
  const float* W2a  = (const float*)d_in[4];
  const float* b2a  = (const float*)d_in[5];
  const float* W1b  = (const float*)d_in[6];
  const float* b1b  = (const float*)d_in[7];
  const float* W2b  = (const float*)d_in[8];
  const float* b2b  = (const float*)d_in[9];
  float* out = (float*)d_out;

  const long long E = in_sizes[1] / 2;
  const int* src = eidx;
  const int* dst = eidx + E;

  // Workspace: buf0 = N*64 (later reused as N*40), buf1 = N*128, buf2 = N*128
  float* buf0 = (float*)d_ws;
  float* buf1 = buf0 + (size_t)NNODES * 64;
  float* buf2 = buf1 + (size_t)NNODES * 128;

  const int MT = NNODES / 16;  // 6250 row tiles, exact

  // ---- conv1: h_pre1 = x + segsum(x[src] -> dst) -------------------------
  hipMemcpyAsync(buf0, x, (size_t)NNODES * 64 * sizeof(float),
                 hipMemcpyDeviceToDevice, stream);
  {
    long long nt = E * 64;
    int blocks = (int)((nt + 255) / 256);
    scatter_add_kernel<6><<<blocks, 256, 0, stream>>>(x, buf0, src, dst, nt);
  }
  // mid = relu(h_pre1 @ W1a + b1a)   [N x 128] -> buf2
  gemm_wmma_f32<64, 128, 1><<<dim3(8, MT), 32, 0, stream>>>(buf0, W1a, b1a, buf2);
  // h1 = relu(mid @ W2a + b2a)       [N x 128] -> buf1   (relu after conv1)
  gemm_wmma_f32<128, 128, 1><<<dim3(8, MT), 32, 0, stream>>>(buf2, W2a, b2a, buf1);

  // ---- conv2: h_pre2 = h1 + segsum(h1[src] -> dst) -----------------------
  hipMemcpyAsync(buf2, buf1, (size_t)NNODES * 128 * sizeof(float),
                 hipMemcpyDeviceToDevice, stream);
  {
    long long nt = E * 128;
    int blocks = (int)((nt + 255) / 256);
    scatter_add_kernel<7><<<blocks, 256, 0, stream>>>(buf1, buf2, src, dst, nt);
  }
  // t = relu(h_pre2 @ W1b + b1b)     [N x 40] -> buf0 (reuse)
  gemm_wmma_f32<128, 40, 1><<<dim3(3, MT), 32, 0, stream>>>(buf2, W1b, b1b, buf0);
  // logits = t @ W2b + b2b           [N x 40] -> buf1 (reuse)
  gemm_wmma_f32<40, 40, 0><<<dim3(3, MT), 32, 0, stream>>>(buf0, W2b, b2b, buf1);

  // ---- log_softmax --------------------------------------------------------
  {
    int blocks = (NNODES * 32 + 255) / 256;
    log_softmax40_kernel<<<blocks, 256, 0, stream>>>(buf1, out, NNODES);
  }
}